// MPSLayer_18262200943503
// MI455X (gfx1250) — compile-verified
//
#include <hip/hip_runtime.h>
#include <hip/hip_bf16.h>

// MPS layer, algebraically collapsed:
//   out[b,:] = (prod_d x[b,d]) * (ones_16 @ C_0 @ ... @ C_{D-1} @ P) + bias
// Kernel 1/2: parallel-associative chain of 16x16 f32 matmuls via
//             v_wmma_f32_16x16x4_f32 (4 chained K=4 WMMAs per step).
// Kernel 3:   chain tail + ones @ M @ P  -> w[64]
// Kernel 4:   HBM-bound streaming product reduction over inputs + rank-1 epilogue.

typedef __attribute__((ext_vector_type(2))) float v2f;
typedef __attribute__((ext_vector_type(8))) float v8f;
typedef __attribute__((ext_vector_type(4))) float fvec4;

// One 16x16 @ 16x16 f32 step: m (LDS, row-major) <- m @ b (LDS, row-major).
// WMMA f32 16x16x4 operand layouts (CDNA5 ISA):
//   A 16x4 : lanes 0-15 hold M=lane, VGPR0/1 = K,K+1; lanes 16-31 shift K by 2
//   B 4x16 : VGPR0 = rows K0 (lanes 0-15) / K0+2 (lanes 16-31); VGPR1 = K0+1 / K0+3
//   C/D    : VGPR r = row r (lanes 0-15) / row r+8 (lanes 16-31)
__device__ __forceinline__ void mat_chain_step(float* m, const float* b, int lo, int hi) {
  v2f a[4], bb[4];
#pragma unroll
  for (int c = 0; c < 4; ++c) {
    int k = 4 * c + 2 * hi;
    a[c].x  = m[lo * 16 + k];
    a[c].y  = m[lo * 16 + k + 1];
    bb[c].x = b[k * 16 + lo];
    bb[c].y = b[(k + 1) * 16 + lo];
  }
  __syncthreads();  // all lanes done reading m before any lane overwrites it
  v8f acc = {};
#pragma unroll
  for (int c = 0; c < 4; ++c) {
    acc = __builtin_amdgcn_wmma_f32_16x16x4_f32(
        false, a[c], false, bb[c], (short)0, acc, false, false);
  }
#pragma unroll
  for (int r = 0; r < 8; ++r) m[(r + 8 * hi) * 16 + lo] = acc[r];
  __syncthreads();
}

// Each block (1 wave, 32 threads) computes the ordered product of n (<=16)
// consecutive 16x16 matrices: dst[g] = src[g*n+0] @ src[g*n+1] @ ... (left->right)
__global__ void mps_chain_kernel(const float* __restrict__ src,
                                 float* __restrict__ dst, int n) {
  __shared__ float mats[16 * 256];
  __shared__ float m[256];
  const int lane = threadIdx.x;
  const int lo = lane & 15, hi = lane >> 4;
  const int total = n * 256;
  const float* gsrc = src + (size_t)blockIdx.x * (size_t)total;
  for (int i = lane; i < total; i += 32) mats[i] = gsrc[i];
  __syncthreads();
#pragma unroll
  for (int k = 0; k < 8; ++k) m[lane + 32 * k] = mats[lane + 32 * k];
  __syncthreads();
  for (int i = 1; i < n; ++i) mat_chain_step(m, &mats[i * 256], lo, hi);
  float* gdst = dst + (size_t)blockIdx.x * 256;
#pragma unroll
  for (int k = 0; k < 8; ++k) gdst[lane + 32 * k] = m[lane + 32 * k];
}

// Single block (1 wave): M = product of n (<=16) matrices, then
// w[o] = sum_r (sum_i M[i][r]) * proj[r*64 + o]   (ones @ M @ P), O = 64.
__global__ void mps_project_kernel(const float* __restrict__ src, int n,
                                   const float* __restrict__ proj,
                                   float* __restrict__ w) {
  __shared__ float mats[16 * 256];
  __shared__ float m[256];
  __shared__ float v[16];
  const int lane = threadIdx.x;
  const int lo = lane & 15, hi = lane >> 4;
  const int total = n * 256;
  for (int i = lane; i < total; i += 32) mats[i] = src[i];
  __syncthreads();
#pragma unroll
  for (int k = 0; k < 8; ++k) m[lane + 32 * k] = mats[lane + 32 * k];
  __syncthreads();
  for (int i = 1; i < n; ++i) mat_chain_step(m, &mats[i * 256], lo, hi);
  if (lane < 16) {
    float s = 0.f;
#pragma unroll
    for (int i = 0; i < 16; ++i) s += m[i * 16 + lane];  // column sums = ones @ M
    v[lane] = s;
  }
  __syncthreads();
  float w0 = 0.f, w1 = 0.f;
#pragma unroll
  for (int r = 0; r < 16; ++r) {
    const float vr = v[r];
    w0 += vr * proj[r * 64 + lane];
    w1 += vr * proj[r * 64 + lane + 32];
  }
  w[lane] = w0;
  w[lane + 32] = w1;
}

// Streaming pass: one wave per batch row. s = prod_d x[row,d]  (float4 NT loads,
// wave32 shuffle-product reduce), then out[row,o] = s*w[o] + bias[o], O = 64.
__global__ void mps_rows_kernel(const float* __restrict__ x,
                                const float* __restrict__ w,
                                const float* __restrict__ bias,
                                float* __restrict__ out, int Dq /* = D/4 */) {
  const int lane = threadIdx.x & 31;
  const int wv = threadIdx.x >> 5;
  const size_t row = (size_t)blockIdx.x * 8 + (size_t)wv;
  const fvec4* rx = (const fvec4*)x + row * (size_t)Dq;
  float p = 1.0f;
  for (int i = lane; i < Dq; i += 32) {
    fvec4 t = __builtin_nontemporal_load(&rx[i]);
    p *= t.x * t.y * t.z * t.w;
  }
#pragma unroll
  for (int off = 16; off; off >>= 1) p *= __shfl_xor(p, off, 32);
  float* orow = out + row * 64;
  orow[lane]      = p * w[lane]      + bias[lane];
  orow[lane + 32] = p * w[lane + 32] + bias[lane + 32];
}

extern "C" void kernel_launch(void* const* d_in, const int* in_sizes, int n_in,
                              void* d_out, int out_size, void* d_ws, size_t ws_size,
                              hipStream_t stream) {
  const float* inputs = (const float*)d_in[0];  // (B, D) f32
  const float* cores  = (const float*)d_in[1];  // (D, 16, 16) f32
  const float* proj   = (const float*)d_in[2];  // (16, 64) f32
  const float* bias   = (const float*)d_in[3];  // (64,) f32

  const int D = in_sizes[1] / 256;          // 2048
  const int B = in_sizes[0] / D;            // 8192

  // Associative 3-level reduction of the matrix chain: 2048 = 128*16 -> 16*8 -> 16
  const int n0 = 16, G0 = D / n0;           // 128 groups of 16
  const int n1 = 8,  G1 = G0 / n1;          // 16 groups of 8
  const int n2 = G1;                        // final chain of 16

  float* ws0  = (float*)d_ws;               // G0 * 256 floats
  float* ws1  = ws0 + (size_t)G0 * 256;     // G1 * 256 floats
  float* wbuf = ws1 + (size_t)G1 * 256;     // 64 floats

  mps_chain_kernel<<<G0, 32, 0, stream>>>(cores, ws0, n0);
  mps_chain_kernel<<<G1, 32, 0, stream>>>(ws0, ws1, n1);
  mps_project_kernel<<<1, 32, 0, stream>>>(ws1, n2, proj, wbuf);
  mps_rows_kernel<<<B / 8, 256, 0, stream>>>(inputs, wbuf, bias, (float*)d_out, D / 4);
}